// GRNN_71502615544225
// MI455X (gfx1250) — compile-verified
//
#include <hip/hip_runtime.h>
#include <hip/hip_bf16.h>

// GRNN fused kernel for gfx1250 (MI455X).
// dist^2 = ||x||^2 + ||t||^2 - 2 x.t  via V_WMMA_F32_16X16X4_F32 (fp32 matrix pipe),
// weights = exp(-dist/2s^2) fused in registers, weights @ tout also via WMMA.
// B tiles staged block-cooperatively through a 2-deep LDS ring (kills the VGPR
// spills of the register double buffer, 4x less global traffic); GEMM2 pipelined
// one iteration behind; v_rcp instead of a per-iteration fdiv expansion.

typedef __attribute__((ext_vector_type(2))) float v2f;
typedef __attribute__((ext_vector_type(8))) float v8f;

#define GB 2048      // batch
#define GN 100000    // train points
#define GD 64        // feature dim
#define GO 16        // output dim
#define NSPLIT 25    // N-splits (25 * 4000 = 100000 exactly)
#define NCHUNK 4000  // n per block
#define ITERS 125    // 125 * 32 = 4000
#define WPAD 20      // padded row stride (words) for LDS weight tile
#define BPAD 68      // padded row stride (words) for LDS B tile (banks 4r+k, 16B-aligned)

// ---------------- kernel 1: row squared norms ----------------
__global__ __launch_bounds__(256) void grnn_sq(const float* __restrict__ tin,
                                               const float* __restrict__ x,
                                               float* __restrict__ tsq,
                                               float* __restrict__ xsq) {
    int i = blockIdx.x * 256 + threadIdx.x;
    if (i < GN) {
        const float4* p = (const float4*)(tin + (size_t)i * GD);
        float s = 0.f;
#pragma unroll
        for (int j = 0; j < GD / 4; ++j) {
            float4 v = p[j];
            s += v.x * v.x + v.y * v.y + v.z * v.z + v.w * v.w;
        }
        tsq[i] = s;
    }
    if (i < GB) {
        const float4* p = (const float4*)(x + (size_t)i * GD);
        float s = 0.f;
#pragma unroll
        for (int j = 0; j < GD / 4; ++j) {
            float4 v = p[j];
            s += v.x * v.x + v.y * v.y + v.z * v.z + v.w * v.w;
        }
        xsq[i] = s;
    }
}

// ---------------- kernel 2: fused main ----------------
// grid: (32 b-tiles, 25 n-splits), block: 256 threads = 8 waves.
// wave w: b_sub = w&3 (16-row subtile), n_sub = w>>2 (16-col subtile). NT=32/iter.
__global__ __launch_bounds__(256, 1) void grnn_main(const float* __restrict__ x,
                                                    const float* __restrict__ tin,
                                                    const float* __restrict__ tout,
                                                    const float* __restrict__ spread,
                                                    const float* __restrict__ xsq,
                                                    const float* __restrict__ tsq,
                                                    float* __restrict__ pout,  // [NSPLIT][GB][GO]
                                                    float* __restrict__ psum)  // [NSPLIT][GB]
{
    __shared__ float btile[2][32 * BPAD];     // 2-deep ring of 32x64 tin tiles (shared by block)
    __shared__ float lds_w[2][8][16 * WPAD];  // 2-deep ring of per-wave weight tiles
    __shared__ float lds_o[8][256];           // per-wave 16x16 output partials
    __shared__ float lds_s[8][16];            // per-wave 16 row sums

    const int tid  = threadIdx.x;
    const int wid  = tid >> 5;
    const int lane = tid & 31;
    const int lo   = lane & 15;   // M (A/D rows) or N (B/D cols)
    const int hi   = lane >> 4;   // K-half selector for A/B; +8 row for C/D

    const int bx    = blockIdx.x;  // 0..31
    const int by    = blockIdx.y;  // 0..24
    const int b_sub = wid & 3;
    const int n_sub = wid >> 2;
    const int b0    = bx * 64 + b_sub * 16;
    const int nbase = by * NCHUNK + n_sub * 16;
    const int gbase = by * NCHUNK;             // block-level n base for staging

    // staging role: thread copies row (tid>>3), floats [8*(tid&7), +8) of the 32x64 tile
    const int st_row = tid >> 3;
    const int st_seg = tid & 7;

    const float sp     = spread[0];
    const float inv2s2 = __builtin_amdgcn_rcpf(2.0f * sp * sp);  // v_rcp_f32, hoistable

    // A-matrix regs: rows b0..b0+15 of x, 16x4 fp32 A-layout per WMMA:
    // lane(0..15|16..31) -> M = lo ; VGPR pair (2i,2i+1) -> K = 4i + 2*hi + {0,1}
    v2f a[16];
    {
        const float* xp = x + (size_t)(b0 + lo) * GD + 2 * hi;
#pragma unroll
        for (int i = 0; i < 16; ++i) a[i] = *(const v2f*)(xp + 4 * i);
    }
    float xsq8[8];
#pragma unroll
    for (int r = 0; r < 8; ++r) xsq8[r] = xsq[b0 + r + 8 * hi];

    v8f oacc = {0.f, 0.f, 0.f, 0.f, 0.f, 0.f, 0.f, 0.f};
    float wsum8[8] = {0.f, 0.f, 0.f, 0.f, 0.f, 0.f, 0.f, 0.f};

    // prologue: stage tile 0 into btile[0]
    {
        const float* gp = tin + (size_t)(gbase + st_row) * GD + st_seg * 8;
        float4 f0 = *(const float4*)(gp);
        float4 f1 = *(const float4*)(gp + 4);
        *(float4*)&btile[0][st_row * BPAD + st_seg * 8]     = f0;
        *(float4*)&btile[0][st_row * BPAD + st_seg * 8 + 4] = f1;
    }
    __syncthreads();

    const float* brow = &btile[0][(n_sub * 16 + lo) * BPAD];  // this lane's B row (buf 0)
    const int    bstep = 32 * BPAD;                           // words between ring buffers

    for (int it = 0; it < ITERS; ++it) {
        const int cur = it & 1;
        const int n0  = nbase + it * 32;

        // issue global loads for the NEXT tile early (latency hidden by WMMA chain)
        float4 f0, f1;
        if (it + 1 < ITERS) {
            const float* gp = tin + (size_t)(gbase + (it + 1) * 32 + st_row) * GD + st_seg * 8;
            f0 = *(const float4*)(gp);
            f1 = *(const float4*)(gp + 4);
        }

        // pipelined GEMM2 operand fetch for the PREVIOUS weight tile (it-1)
        v2f wa[4], tb[4];
        if (it > 0) {
            const int pn0 = n0 - 32;
#pragma unroll
            for (int i = 0; i < 4; ++i) {
                const int k = 4 * i + 2 * hi;
                wa[i]  = *(const v2f*)&lds_w[cur ^ 1][wid][lo * WPAD + k];
                tb[i].x = tout[(size_t)(pn0 + k) * GO + lo];
                tb[i].y = tout[(size_t)(pn0 + k + 1) * GO + lo];
            }
        }

        // GEMM1: cross[m][n] = dot(x[b0+m], tin[n0+n]); two interleaved C chains.
        // B operands read just-in-time from the LDS ring (short ds latency).
        const float* bp = brow + cur * bstep + 2 * hi;
        v8f c0 = {0.f, 0.f, 0.f, 0.f, 0.f, 0.f, 0.f, 0.f};
        v8f c1 = {0.f, 0.f, 0.f, 0.f, 0.f, 0.f, 0.f, 0.f};
#pragma unroll
        for (int i = 0; i < 16; i += 2) {
            v2f bv0 = *(const v2f*)(bp + 4 * i);
            v2f bv1 = *(const v2f*)(bp + 4 * (i + 1));
            c0 = __builtin_amdgcn_wmma_f32_16x16x4_f32(false, a[i], false, bv0,
                                                       (short)0, c0, false, false);
            c1 = __builtin_amdgcn_wmma_f32_16x16x4_f32(false, a[i + 1], false, bv1,
                                                       (short)0, c1, false, false);
        }

        // pipelined GEMM2 for previous tile: oacc[m][o] += sum_k W[m][k]*tout[pn0+k][o]
        if (it > 0) {
#pragma unroll
            for (int i = 0; i < 4; ++i)
                oacc = __builtin_amdgcn_wmma_f32_16x16x4_f32(false, wa[i], false, tb[i],
                                                             (short)0, oacc, false, false);
        }

        // fuse: dist -> weight; D layout: VGPR r, lane -> (M = r + 8*hi, N = lo)
        const float tsq_l = tsq[n0 + lo];
#pragma unroll
        for (int r = 0; r < 8; ++r) {
            float cr   = c0[r] + c1[r];
            float dist = xsq8[r] + tsq_l - 2.0f * cr;
            float w    = __expf(-dist * inv2s2);
            wsum8[r] += w;
            lds_w[cur][wid][(r + 8 * hi) * WPAD + lo] = w;  // b-major [m][k=n_local]
        }

        // stage the prefetched tile into the other ring buffer; one barrier/iter:
        // writes to buf cur^1 only start after everyone's reads of it (prev iter)
        // and this iteration's reads of buf cur are all before the barrier.
        if (it + 1 < ITERS) {
            float* dp = &btile[cur ^ 1][st_row * BPAD + st_seg * 8];
            *(float4*)(dp)     = f0;
            *(float4*)(dp + 4) = f1;
        }
        __syncthreads();
    }

    // epilogue: GEMM2 for the last weight tile
    {
        const int pn0 = nbase + (ITERS - 1) * 32;
        const int buf = (ITERS - 1) & 1;
#pragma unroll
        for (int i = 0; i < 4; ++i) {
            const int k = 4 * i + 2 * hi;
            v2f wa = *(const v2f*)&lds_w[buf][wid][lo * WPAD + k];
            v2f tb;
            tb.x = tout[(size_t)(pn0 + k) * GO + lo];
            tb.y = tout[(size_t)(pn0 + k + 1) * GO + lo];
            oacc = __builtin_amdgcn_wmma_f32_16x16x4_f32(false, wa, false, tb,
                                                         (short)0, oacc, false, false);
        }
    }

    // reduce row sums across the 16 lanes of each half (cols of the tile)
#pragma unroll
    for (int r = 0; r < 8; ++r) {
        float v = wsum8[r];
        v += __shfl_xor(v, 1, 32);
        v += __shfl_xor(v, 2, 32);
        v += __shfl_xor(v, 4, 32);
        v += __shfl_xor(v, 8, 32);
        wsum8[r] = v;
    }
    if (lo == 0) {
#pragma unroll
        for (int r = 0; r < 8; ++r) lds_s[wid][r + 8 * hi] = wsum8[r];
    }
#pragma unroll
    for (int r = 0; r < 8; ++r) lds_o[wid][(r + 8 * hi) * 16 + lo] = oacc[r];

    __syncthreads();

    // combine the two n_sub halves, write deterministic per-block partials
    for (int idx = tid; idx < 1024; idx += 256) {
        int bl = idx >> 4;  // 0..63 local b row
        int o  = idx & 15;
        int bs = bl >> 4;   // b_sub
        int m  = bl & 15;
        float v = lds_o[bs][m * 16 + o] + lds_o[bs + 4][m * 16 + o];
        pout[((size_t)by * GB + bx * 64 + bl) * GO + o] = v;
    }
    if (tid < 64) {
        int bs = tid >> 4, m = tid & 15;
        psum[(size_t)by * GB + bx * 64 + tid] = lds_s[bs][m] + lds_s[bs + 4][m];
    }
}

// ---------------- kernel 3: final reduction + divide ----------------
__global__ __launch_bounds__(256) void grnn_reduce(const float* __restrict__ pout,
                                                   const float* __restrict__ psum,
                                                   float* __restrict__ out) {
    int idx = blockIdx.x * 256 + threadIdx.x;  // 0..GB*GO-1
    int b = idx >> 4;
    int o = idx & 15;
    float acc = 0.f, s = 0.f;
#pragma unroll 5
    for (int by = 0; by < NSPLIT; ++by) {
        acc += pout[((size_t)by * GB + b) * GO + o];
        s   += psum[(size_t)by * GB + b];
    }
    out[idx] = acc / s;
}

extern "C" void kernel_launch(void* const* d_in, const int* in_sizes, int n_in,
                              void* d_out, int out_size, void* d_ws, size_t ws_size,
                              hipStream_t stream) {
    const float* x      = (const float*)d_in[0];  // [2048,64]
    const float* tin    = (const float*)d_in[1];  // [100000,64]
    const float* tout   = (const float*)d_in[2];  // [100000,16]
    const float* spread = (const float*)d_in[3];  // [1]

    float* ws   = (float*)d_ws;
    float* tsq  = ws;                       // 100000 floats
    float* xsq  = ws + 100352;              // 2048 floats
    float* pout = ws + 102400;              // NSPLIT*GB*GO = 819200 floats
    float* psum = ws + 102400 + 819200;     // NSPLIT*GB    = 51200 floats

    grnn_sq<<<(GN + 255) / 256, 256, 0, stream>>>(tin, x, tsq, xsq);
    grnn_main<<<dim3(GB / 64, NSPLIT), 256, 0, stream>>>(x, tin, tout, spread,
                                                         xsq, tsq, pout, psum);
    grnn_reduce<<<(GB * GO) / 256, 256, 0, stream>>>(pout, psum, (float*)d_out);
}